// DCR_21285857919673
// MI455X (gfx1250) — compile-verified
//
#include <hip/hip_runtime.h>

typedef __attribute__((ext_vector_type(2))) float v2f;
typedef __attribute__((ext_vector_type(8))) float v8f;

#define NEG_INF (-10000.0f)
#define EPS_F   (1e-8f)

constexpr int B = 32, S = 1024, H = 1024, L = 32;
constexpr int KWAVES = 4;            // K-split factor (waves per tile)
constexpr int KSLICE = H / KWAVES;   // 256 K-values per wave

// ---------------------------------------------------------------------------
// Phase 1: one streaming pass over seq (128 MB read once -> ~5.5us HBM floor).
// 128-thread blocks = 4 waves per 16-row tile; each wave owns a 256-wide
// K-slice (4x occupancy vs one-wave-per-tile, to hide HBM latency).
// d1/d2 via V_WMMA_F32_16X16X4_F32 with q1,q2 in B-matrix columns 0,1;
// n2 via per-lane FMA on the same A tile + shfl_xor(16) (wave32).
// Partials (16 x {d1,d2,n2} per wave) reduced deterministically in LDS.
// ---------------------------------------------------------------------------
__global__ __launch_bounds__(128) void phase1_wmma(
    const float* __restrict__ seq, const int* __restrict__ idxs,
    float* __restrict__ d1, float* __restrict__ d2, float* __restrict__ n2)
{
  const int tile    = blockIdx.x;          // 0 .. B*(S/16)-1
  const int b       = tile >> 6;           // tile / (S/16)
  const int rowBase = (tile & 63) << 4;
  const int tid     = threadIdx.x;
  const int wave    = tid >> 5;            // 0..3 : K-slice owner
  const int lane    = tid & 31;            // wave32 lane
  const int half    = lane >> 4;           // K-half selector in A/B layout
  const int lm      = lane & 15;           // row (A) / column (B) index

  const int sep0 = idxs[2 * b];
  const int k0   = wave * KSLICE;

  // A: lane lm streams row (rowBase+lm); contiguous K, halves offset by 2.
  const float* rowA = seq + ((size_t)b * S + rowBase + lm) * H + k0 + 2 * half;
  // B: column 0 = q1, column 1 = q2, columns 2..15 forced to zero via select.
  const float* q1p = seq + ((size_t)b * S + 1) * H + k0 + 2 * half;
  const float* q2p = seq + ((size_t)b * S + (sep0 - 1)) * H + k0 + 2 * half;
  const float* qp  = (lm == 1) ? q2p : q1p;    // every lane: valid pointer
  const bool qlane = (lm < 2);                 // columns 2..15 -> zero

  v8f   c    = {};
  float nacc = 0.0f;

#pragma unroll 8
  for (int k = 0; k < KSLICE; k += 4) {
    const float a0 = rowA[k];
    const float a1 = rowA[k + 1];
    const float t0 = qp[k];                    // unconditional load,
    const float t1 = qp[k + 1];                //  lanes 2..15 hit L0/L2
    const float b0 = qlane ? t0 : 0.0f;        // v_cndmask, no EXEC churn
    const float b1 = qlane ? t1 : 0.0f;

    nacc = fmaf(a0, a0, fmaf(a1, a1, nacc));   // self-dot for n2

    v2f av = {a0, a1};
    v2f bv = {b0, b1};
    c = __builtin_amdgcn_wmma_f32_16x16x4_f32(
            /*neg_a=*/false, av, /*neg_b=*/false, bv,
            /*c_mod=*/(short)0, c, /*reuse_a=*/false, /*reuse_b=*/false);
  }

  // Deterministic cross-wave reduction of 16-row partials through LDS.
  __shared__ float pd1[KWAVES][16];
  __shared__ float pd2[KWAVES][16];
  __shared__ float pn2[KWAVES][16];

  // n2: lane m covers K%4 in {0,1}, lane m+16 covers {2,3} -> combine halves.
  const float nrow = nacc + __shfl_xor(nacc, 16, 32);
  if (lane < 16) pn2[wave][lane] = nrow;

  // C/D layout: VGPR r, lanes 0-15 -> M=r, N=lane; lanes 16-31 -> M=r+8.
  // d1 = column 0 (lanes 0,16); d2 = column 1 (lanes 1,17).
#pragma unroll
  for (int r = 0; r < 8; ++r) {
    const float v = c[r];
    if (lm == 0) pd1[wave][half * 8 + r] = v;
    if (lm == 1) pd2[wave][half * 8 + r] = v;
  }
  __syncthreads();

  if (tid < 16) {
    float s1 = 0.0f, s2 = 0.0f, sn = 0.0f;
#pragma unroll
    for (int w = 0; w < KWAVES; ++w) {
      s1 += pd1[w][tid];
      s2 += pd2[w][tid];
      sn += pn2[w][tid];
    }
    const size_t o = (size_t)b * S + rowBase + tid;
    d1[o] = s1;
    d2[o] = s2;
    n2[o] = sn;
  }
}

// ---------------------------------------------------------------------------
// Phase 2: windowed argmax scan. 256 i-positions per block; stage the
// 256+L sliding window of d2/n2 in LDS. Strict '>' preserves first-tie argmax.
// ---------------------------------------------------------------------------
__global__ __launch_bounds__(256) void phase2_scan(
    const float* __restrict__ d1, const float* __restrict__ d2,
    const float* __restrict__ n2, const int* __restrict__ idxs,
    float* __restrict__ maxv, int* __restrict__ endi)
{
  __shared__ float d2s[256 + L];
  __shared__ float n2s[256 + L];

  const int blocksPerBatch = S / 256;                 // 4
  const int b     = blockIdx.x / blocksPerBatch;
  const int iBase = (blockIdx.x % blocksPerBatch) * 256;
  const int tid   = threadIdx.x;
  const size_t base = (size_t)b * S;

  for (int t = tid; t < 256 + L; t += 256) {
    const int j = min(iBase + t, S - 1);              // j = min(i+l, S-1)
    d2s[t] = d2[base + j];
    n2s[t] = n2[base + j];
  }
  __syncthreads();

  const int sep0 = idxs[2 * b];
  const int sep1 = idxs[2 * b + 1];
  const float qn2 = n2[base + 1] + n2[base + sep0 - 1];

  const int   i   = iBase + tid;
  const float d1i = d1[base + i];
  const float n2i = n2[base + i];

  float best  = -__builtin_inff();
  int   bestl = 0;
#pragma unroll
  for (int l = 0; l < L; ++l) {
    const float num = d1i + d2s[tid + l];
    const float den = fmaxf(sqrtf((n2i + n2s[tid + l]) * qn2), EPS_F);
    const float sim = ((i + l) < sep1) ? (num / den) : NEG_INF;
    if (sim > best) { best = sim; bestl = l; }
  }

  const bool valid_i = (i > sep0) && (i < sep1);
  maxv[base + i] = valid_i ? best : NEG_INF;
  endi[base + i] = valid_i ? (i + bestl) : -1;
}

// ---------------------------------------------------------------------------
extern "C" void kernel_launch(void* const* d_in, const int* in_sizes, int n_in,
                              void* d_out, int out_size, void* d_ws, size_t ws_size,
                              hipStream_t stream) {
  const float* seq  = (const float*)d_in[0];
  const int*   idxs = (const int*)d_in[1];
  // d_in[2] = max_ans_len (compile-time constant L=32 per reference setup)

  float* d1 = (float*)d_ws;                 // B*S floats
  float* d2 = d1 + (size_t)B * S;           // B*S floats
  float* n2 = d2 + (size_t)B * S;           // B*S floats

  phase1_wmma<<<B * (S / 16), 128, 0, stream>>>(seq, idxs, d1, d2, n2);

  float* maxv = (float*)d_out;              // outputs concatenated flat:
  int*   endi = (int*)d_out + (size_t)B * S;//   [max_vals fp32 | end_idx i32]

  phase2_scan<<<(B * S) / 256, 256, 0, stream>>>(d1, d2, n2, idxs, maxv, endi);
}